// LapLoss_1211180777949
// MI455X (gfx1250) — compile-verified
//
#include <hip/hip_runtime.h>

typedef float v2f __attribute__((ext_vector_type(2)));
typedef float v8f __attribute__((ext_vector_type(8)));

// jnp.pad(..., mode='reflect') index mapping for pad <= 2, n >= 3
__device__ __forceinline__ int refl(int i, int n) {
  i = (i < 0) ? -i : i;
  i = (i >= n) ? (2 * n - 2 - i) : i;
  return i;
}

// Separable Gaussian taps {1,4,6,4,1}/16 ; 2D kernel = outer product (sum 1).
__device__ __forceinline__ float gw(int i) {
  const float t[5] = {1.f / 16.f, 4.f / 16.f, 6.f / 16.f, 4.f / 16.f, 1.f / 16.f};
  return t[i];
}

// down = conv5x5_reflect(cur)[::2, ::2], fused for both tensors.
__global__ __launch_bounds__(256) void down_kernel(
    const float* __restrict__ cx, const float* __restrict__ cy,
    float* __restrict__ dx, float* __restrict__ dy,
    int h, int w, int total) {
  int h2 = h >> 1, w2 = w >> 1;
  int p = blockIdx.x * blockDim.x + threadIdx.x;
  if (p >= total) return;
  int j2 = p % w2;
  int i2 = (p / w2) % h2;
  int img = p / (w2 * h2);
  const long base = (long)img * h * w;
  float ax = 0.f, ay = 0.f;
#pragma unroll
  for (int i = 0; i < 5; ++i) {
    int sy = refl(2 * i2 + i - 2, h);
    const float* rx = cx + base + (long)sy * w;
    const float* ry = cy + base + (long)sy * w;
    float sxv = 0.f, syv = 0.f;
#pragma unroll
    for (int j = 0; j < 5; ++j) {
      int sx = refl(2 * j2 + j - 2, w);
      sxv += gw(j) * rx[sx];
      syv += gw(j) * ry[sx];
    }
    ax += gw(i) * sxv;
    ay += gw(i) * syv;
  }
  dx[p] = ax;
  dy[p] = ay;
}

// Polyphase upsample (zero-stuff + 4*conv5x5, reflect on the stuffed grid),
// Laplacian diff |(cx-upx)-(cy-upy)|, wave reduction via V_WMMA_F32_16X16X4_F32,
// block sum -> f64 global atomic accumulator for this level.
__global__ __launch_bounds__(256) void lap_kernel(
    const float* __restrict__ cx, const float* __restrict__ cy,
    const float* __restrict__ dx, const float* __restrict__ dy,
    double* __restrict__ acc, int h, int w, int total) {
  int h2 = h >> 1, w2 = w >> 1;
  int p0 = blockIdx.x * blockDim.x + threadIdx.x;
  bool valid = (p0 < total);
  int p = valid ? p0 : (total - 1);   // clamp: keep EXEC all-ones for WMMA
  int x = p % w;
  int y = (p / w) % h;
  int img = p / (w * h);
  const long base2 = (long)img * h2 * w2;

  float ux = 0.f, uy = 0.f;
#pragma unroll
  for (int i = 0; i < 5; ++i) {
    int zy = refl(y + i - 2, h);      // index on zero-stuffed grid (size h)
    if ((zy & 1) == 0) {              // odd rows of stuffed grid are zero
      const float* rx = dx + base2 + (long)(zy >> 1) * w2;
      const float* ry = dy + base2 + (long)(zy >> 1) * w2;
      float sxv = 0.f, syv = 0.f;
#pragma unroll
      for (int j = 0; j < 5; ++j) {
        int zx = refl(x + j - 2, w);
        if ((zx & 1) == 0) {
          sxv += gw(j) * rx[zx >> 1];
          syv += gw(j) * ry[zx >> 1];
        }
      }
      ux += gw(i) * sxv;
      uy += gw(i) * syv;
    }
  }
  ux *= 4.f;                          // upsample kernel is 4 * gauss
  uy *= 4.f;
  float t = valid ? fabsf((cx[p] - ux) - (cy[p] - uy)) : 0.f;

  // --- wave32 reduction with one fp32 WMMA ---
  // A (16x4): lanes 0-15 hold A[m][0]=t_m (K0 in vgpr0), lanes 16-31 hold
  // A[m][2]=t_{m+16}; K1/K3 zero. B = ones(4x16).
  // D[m][n] = t_m + t_{m+16}; summing a lane's 8 acc regs gives a half-sum,
  // one shfl_xor(16) completes the 32-lane sum.
  v2f a; a.x = t;   a.y = 0.f;
  v2f b; b.x = 1.f; b.y = 1.f;
  v8f c = {};
  c = __builtin_amdgcn_wmma_f32_16x16x4_f32(
      /*neg_a=*/false, a, /*neg_b=*/false, b,
      /*c_mod=*/(short)0, c, /*reuse_a=*/false, /*reuse_b=*/false);
  float s = c[0] + c[1] + c[2] + c[3] + c[4] + c[5] + c[6] + c[7];
  s += __shfl_xor(s, 16, 32);

  __shared__ float wsum[8];
  int lane = threadIdx.x & 31;
  int wid = threadIdx.x >> 5;
  if (lane == 0) wsum[wid] = s;
  __syncthreads();
  if (threadIdx.x == 0) {
    float bs = 0.f;
    int nw = (blockDim.x + 31) >> 5;
    for (int i = 0; i < nw; ++i) bs += wsum[i];
    atomicAdd(acc, (double)bs);       // global_atomic_add_f64
  }
}

__global__ void finalize_kernel(const double* __restrict__ acc,
                                float* __restrict__ out, int nimg, int h0) {
  if (threadIdx.x == 0 && blockIdx.x == 0) {
    double loss = 0.0;
    for (int l = 0; l < 5; ++l) {
      int h = h0 >> l;
      loss += acc[l] / ((double)nimg * (double)h * (double)h);
    }
    out[0] = (float)loss;
  }
}

extern "C" void kernel_launch(void* const* d_in, const int* in_sizes, int n_in,
                              void* d_out, int out_size, void* d_ws, size_t ws_size,
                              hipStream_t stream) {
  (void)n_in; (void)out_size; (void)ws_size;
  const float* in_x = (const float*)d_in[0];
  const float* in_y = (const float*)d_in[1];
  const int H = 512;                       // reference: (2,16,3,512,512)
  const int NIMG = in_sizes[0] / (H * H);  // 96 planes

  // Workspace layout: [8 x f64 accumulators][down_x/down_y pyramid buffers]
  double* acc = (double*)d_ws;
  char* base = (char*)d_ws + 64;
  float* dnx[5]; float* dny[5];
  size_t off = 0;
  int h = H;
  for (int l = 0; l < 5; ++l) {
    int h2 = h >> 1;
    size_t n = (size_t)NIMG * h2 * h2;
    dnx[l] = (float*)(base + off); off += n * sizeof(float);
    dny[l] = (float*)(base + off); off += n * sizeof(float);
    h = h2;
  }

  hipMemsetAsync(d_ws, 0, 64, stream);     // zero the f64 accumulators

  const float* cx = in_x;
  const float* cy = in_y;
  h = H;
  for (int l = 0; l < 5; ++l) {
    int h2 = h >> 1;
    int totalD = NIMG * h2 * h2;
    int totalL = NIMG * h * h;
    down_kernel<<<(totalD + 255) / 256, 256, 0, stream>>>(
        cx, cy, dnx[l], dny[l], h, h, totalD);
    lap_kernel<<<(totalL + 255) / 256, 256, 0, stream>>>(
        cx, cy, dnx[l], dny[l], acc + l, h, h, totalL);
    cx = dnx[l];
    cy = dny[l];
    h = h2;
  }
  finalize_kernel<<<1, 32, 0, stream>>>(acc, (float*)d_out, NIMG, H);
}